// MyRNN_55911884260052
// MI455X (gfx1250) — compile-verified
//
#include <hip/hip_runtime.h>
#include <hip/hip_bf16.h>

// Problem constants (from reference)
#define BATCH 128
#define SEQ   200
#define ELEN  100
#define UNITS 256
#define U3    768   // 3*UNITS

typedef __attribute__((ext_vector_type(2))) float v2f;
typedef __attribute__((ext_vector_type(8))) float v8f;

// ---------------------------------------------------------------------------
// Phase 1: MX[t, b, 0:768] = emb[inputs[b,t]] @ kernel + bias_i
// One block per (t, 8-Ntile group). x_t gathered into LDS once; kernel columns
// held in registers across the 8 batch tiles. f32 WMMA 16x16x4, K=100=25*4.
// ---------------------------------------------------------------------------
__global__ void __launch_bounds__(256) mx_kernel(
    const int*   __restrict__ inputs,
    const float* __restrict__ emb,
    const float* __restrict__ kern,
    const float* __restrict__ bias_i,
    float*       __restrict__ MX)
{
    __shared__ float xs[BATCH * ELEN];   // 51.2 KB
    __shared__ int   sidx[BATCH];

    const int t   = blockIdx.x;
    const int tid = threadIdx.x;

    if (tid < BATCH) sidx[tid] = inputs[tid * SEQ + t];
    __syncthreads();

    // Gather embedding rows (each row = 25 float4, 16B aligned: 400B rows)
    for (int j = tid; j < BATCH * (ELEN / 4); j += 256) {
        int row = j / (ELEN / 4), q = j % (ELEN / 4);
        const float4* src = (const float4*)(emb + (size_t)sidx[row] * ELEN);
        ((float4*)(xs + row * ELEN))[q] = src[q];
    }
    __syncthreads();

    const int wave  = tid >> 5;                    // 0..7
    const int lane  = tid & 31;
    const int ntile = blockIdx.y * 8 + wave;       // 0..47
    const int n     = ntile * 16 + (lane & 15);    // global column
    const int koff  = (lane < 16) ? 0 : 2;
    const int m_lo  = lane & 15;

    // B (kernel) columns for this ntile: 25 k-steps resident in registers
    v2f breg[25];
#pragma unroll
    for (int ks = 0; ks < 25; ++ks) {
        int k = ks * 4 + koff;
        breg[ks].x = kern[(size_t)k * U3 + n];
        breg[ks].y = kern[(size_t)(k + 1) * U3 + n];
    }
    const float bi  = bias_i[n];
    float*      mxt = MX + (size_t)t * BATCH * U3;

    for (int mt = 0; mt < 8; ++mt) {
        v8f acc = {};
        const float* xrow = xs + (mt * 16 + m_lo) * ELEN;
#pragma unroll
        for (int ks = 0; ks < 25; ++ks) {
            v2f a;
            a.x = xrow[ks * 4 + koff];
            a.y = xrow[ks * 4 + koff + 1];
            acc = __builtin_amdgcn_wmma_f32_16x16x4_f32(
                false, a, false, breg[ks], (short)0, acc, false, false);
        }
#pragma unroll
        for (int j = 0; j < 8; ++j) {
            int m = mt * 16 + j + ((lane < 16) ? 0 : 8);
            mxt[(size_t)m * U3 + n] = acc[j] + bi;
        }
    }
}

// ---------------------------------------------------------------------------
// h := 0
// ---------------------------------------------------------------------------
__global__ void zero_h_kernel(float* __restrict__ h)
{
    h[blockIdx.x * 256 + threadIdx.x] = 0.0f;
}

// ---------------------------------------------------------------------------
// Phase 2 (per time step): MH = h @ rkernel + bias_r, then GRU gate update.
// 16 blocks, block b owns units [16b, 16b+16): stages its 48 rkernel columns
// (z/r/h gates) into LDS packed as [gate][kq][n][kk] so B pairs are ds_b64.
// Wave w computes batch tile w (8 tiles x 16 rows = 128). K=256 = 64*4.
// ---------------------------------------------------------------------------
__global__ void __launch_bounds__(256) gru_step_kernel(
    const float* __restrict__ hin,
    float*       __restrict__ hout,
    const float* __restrict__ rkern,
    const float* __restrict__ bias_r,
    const float* __restrict__ MX,
    int t)
{
    __shared__ float sB[3 * 64 * 16 * 4];   // 48 KB: [gate][kq][n][kk]

    const int tid = threadIdx.x;
    const int ut  = blockIdx.x;             // unit tile 0..15
    const int u0  = ut * 16;

    // Stage rkernel slice: one float4 (4 consecutive k) per (gate,kq,n)
    for (int j = tid; j < 3 * 64 * 16; j += 256) {
        int n  = j & 15;
        int kq = (j >> 4) & 63;
        int g  = j >> 10;
        int col = g * UNITS + u0 + n;
        float4 v;
        v.x = rkern[(size_t)(kq * 4 + 0) * U3 + col];
        v.y = rkern[(size_t)(kq * 4 + 1) * U3 + col];
        v.z = rkern[(size_t)(kq * 4 + 2) * U3 + col];
        v.w = rkern[(size_t)(kq * 4 + 3) * U3 + col];
        ((float4*)sB)[j] = v;
    }
    __syncthreads();

    const int wave   = tid >> 5;            // batch tile 0..7
    const int lane   = tid & 31;
    const int nloc   = lane & 15;
    const int koff   = (lane < 16) ? 0 : 2;
    const int m_base = wave * 16;
    const float* hrow = hin + (size_t)(m_base + (lane & 15)) * UNITS;

    v8f accz = {}, accr = {}, acch = {};
#pragma unroll 8
    for (int kq = 0; kq < 64; ++kq) {
        v2f a;
        a.x = hrow[kq * 4 + koff];
        a.y = hrow[kq * 4 + koff + 1];
        v2f b0 = *(const v2f*)(sB + (((0 * 64 + kq) * 16 + nloc) << 2) + koff);
        v2f b1 = *(const v2f*)(sB + (((1 * 64 + kq) * 16 + nloc) << 2) + koff);
        v2f b2 = *(const v2f*)(sB + (((2 * 64 + kq) * 16 + nloc) << 2) + koff);
        accz = __builtin_amdgcn_wmma_f32_16x16x4_f32(false, a, false, b0, (short)0, accz, false, false);
        accr = __builtin_amdgcn_wmma_f32_16x16x4_f32(false, a, false, b1, (short)0, accr, false, false);
        acch = __builtin_amdgcn_wmma_f32_16x16x4_f32(false, a, false, b2, (short)0, acch, false, false);
    }

    const int   u   = u0 + nloc;
    const float brz = bias_r[u];
    const float brr = bias_r[UNITS + u];
    const float brh = bias_r[2 * UNITS + u];
    const float* mxt = MX + (size_t)t * BATCH * U3;

#pragma unroll
    for (int j = 0; j < 8; ++j) {
        int   m    = m_base + j + ((lane < 16) ? 0 : 8);
        float mhz  = accz[j] + brz;
        float mhr  = accr[j] + brr;
        float mhh  = acch[j] + brh;
        const float* mxrow = mxt + (size_t)m * U3;
        float z    = 1.0f / (1.0f + __expf(-(mxrow[u] + mhz)));
        float r    = 1.0f / (1.0f + __expf(-(mxrow[UNITS + u] + mhr)));
        float cand = tanhf(mxrow[2 * UNITS + u] + r * mhh);
        float hold = hin[(size_t)m * UNITS + u];
        hout[(size_t)m * UNITS + u] = z * hold + (1.0f - z) * cand;
    }
}

// ---------------------------------------------------------------------------
// Phase 3: out[b] = sigmoid(h[b,:] . w_out + b_out)
// ---------------------------------------------------------------------------
__global__ void __launch_bounds__(128) out_kernel(
    const float* __restrict__ h,
    const float* __restrict__ w,
    const float* __restrict__ b,
    float*       __restrict__ out)
{
    int bi = threadIdx.x;   // 0..127
    float s = 0.0f;
#pragma unroll 8
    for (int u = 0; u < UNITS; ++u) s += h[(size_t)bi * UNITS + u] * w[u];
    s += b[0];
    out[bi] = 1.0f / (1.0f + __expf(-s));
}

// ---------------------------------------------------------------------------
extern "C" void kernel_launch(void* const* d_in, const int* in_sizes, int n_in,
                              void* d_out, int out_size, void* d_ws, size_t ws_size,
                              hipStream_t stream)
{
    const int*   inputs = (const int*)  d_in[0];
    const float* emb    = (const float*)d_in[1];
    const float* kern   = (const float*)d_in[2];
    const float* rkern  = (const float*)d_in[3];
    const float* bias_i = (const float*)d_in[4];
    const float* bias_r = (const float*)d_in[5];
    const float* w_out  = (const float*)d_in[6];
    const float* b_out  = (const float*)d_in[7];
    float* out = (float*)d_out;

    char*  ws = (char*)d_ws;
    float* MX = (float*)ws;                                   // 200*128*768 f32 = 78.6 MB
    float* hA = (float*)(ws + (size_t)SEQ * BATCH * U3 * sizeof(float));
    float* hB = hA + BATCH * UNITS;

    // Phase 1: all input-side GEMMs at once (fully parallel)
    mx_kernel<<<dim3(SEQ, 6), 256, 0, stream>>>(inputs, emb, kern, bias_i, MX);

    // h := 0
    zero_h_kernel<<<dim3((BATCH * UNITS) / 256), 256, 0, stream>>>(hA);

    // Phase 2: 200 dependent steps; stream order provides the global sync
    float* hc = hA;
    float* hn = hB;
    for (int t = 0; t < SEQ; ++t) {
        gru_step_kernel<<<dim3(16), 256, 0, stream>>>(hc, hn, rkern, bias_r, MX, t);
        float* tmp = hc; hc = hn; hn = tmp;
    }

    // Phase 3: output head
    out_kernel<<<1, 128, 0, stream>>>(hc, w_out, b_out, out);
}